// BiLSTM_CRF_63625645522968
// MI455X (gfx1250) — compile-verified
//
#include <hip/hip_runtime.h>

typedef __attribute__((ext_vector_type(2))) float v2f;
typedef __attribute__((ext_vector_type(8))) float v8f;

constexpr int SEQ   = 512;   // T
constexpr int BATCH = 32;    // B
constexpr int EMB   = 256;   // E
constexpr int HID   = 128;   // Hh
constexpr int GATES = 512;   // 4*Hh
constexpr int NTAG  = 12;    // K
constexpr int TAG_START = 9;
constexpr int TAG_STOP  = 10;
constexpr float NEGV = -10000.0f;

__device__ __forceinline__ float sigm(float x) { return 1.0f / (1.0f + __expf(-x)); }

// ---------------------------------------------------------------------------
// Kernel A: embedding gather + input-projection GEMM (both directions)
// xg[dir][t*32+b][g] = embed[sentence[b][t]] . W_ih[dir][g] + bias[dir][g]
// One wave computes one 16x16 tile via 64x v_wmma_f32_16x16x4_f32.
// ---------------------------------------------------------------------------
__global__ __launch_bounds__(256) void k_input_gemm(
    const int* __restrict__ sentence, const float* __restrict__ embed,
    const float* __restrict__ Wf, const float* __restrict__ bf,
    const float* __restrict__ Wr, const float* __restrict__ br,
    float* __restrict__ xg)
{
  int wave  = (blockIdx.x * blockDim.x + threadIdx.x) >> 5;   // 0..65535
  int lane  = threadIdx.x & 31;
  int ntile = wave & 31;            // 32 N tiles (512 gates)
  int mtile = (wave >> 5) & 1023;   // 1024 M tiles (16384 rows)
  int dir   = wave >> 15;           // 2 directions
  const float* W    = dir ? Wr : Wf;
  const float* bias = dir ? br : bf;

  int lm   = lane & 15;
  int koff = (lane >> 4) << 1;      // lanes 16..31 hold K+2,K+3
  int row  = mtile * 16 + lm;       // row = t*32 + b
  int t = row >> 5, b = row & 31;
  int tok = sentence[b * SEQ + t];
  const float* arow = embed + (long long)tok * EMB;
  int n = ntile * 16 + lm;
  const float* brow = W + (size_t)n * EMB;

  v8f acc = (v8f)0.0f;
  for (int k = 0; k < EMB; k += 4) {
    v2f a  = *(const v2f*)(arow + k + koff);
    v2f bv = *(const v2f*)(brow + k + koff);
    acc = __builtin_amdgcn_wmma_f32_16x16x4_f32(false, a, false, bv,
                                                (short)0, acc, false, false);
  }
  float bval = bias[n];
  float* outp = xg + (size_t)dir * SEQ * BATCH * GATES;
  int rbase = mtile * 16 + ((lane >> 4) << 3);   // C/D: M = i + 8*(lane>=16)
#pragma unroll
  for (int i = 0; i < 8; ++i)
    outp[(size_t)(rbase + i) * GATES + n] = acc[i] + bval;
}

// ---------------------------------------------------------------------------
// Kernel B: LSTM recurrence, one workgroup per direction (2 blocks x 1024).
// h lives in LDS (padded), gates staged in LDS, c in registers.
// Per step: 32 waves x 2 tiles x 32 WMMA (f32 16x16x4), 2 barriers.
// dir==1 processes xg[T-1-t] and writes hs[T-1-t] (reverse + re-reverse).
// ---------------------------------------------------------------------------
__global__ __launch_bounds__(1024) void k_lstm(
    const float* __restrict__ Whf, const float* __restrict__ Whr,
    const float* __restrict__ h0, const float* __restrict__ c0,
    const float* __restrict__ xg, float* __restrict__ hs)
{
  __shared__ float h_lds[BATCH][HID + 4];   // pad stride 132 -> conflict-free A loads
  __shared__ float g_lds[BATCH][GATES];     // 64 KB gate staging

  int dir = blockIdx.x;
  const float* Wh  = dir ? Whr : Whf;
  const float* xgd = xg + (size_t)dir * SEQ * BATCH * GATES;
  int tid  = threadIdx.x;
  int lane = tid & 31;
  int wave = tid >> 5;                       // 0..31 -> N tile

  for (int p = tid; p < BATCH * HID; p += 1024) {
    int b = p >> 7, j = p & 127;
    h_lds[b][j] = h0[dir * BATCH * HID + p];
  }
  float creg[4];
#pragma unroll
  for (int r = 0; r < 4; ++r)
    creg[r] = c0[dir * BATCH * HID + tid + 1024 * r];
  __syncthreads();

  int lm   = lane & 15;
  int koff = (lane >> 4) << 1;
  int n = wave * 16 + lm;
  const float* brow = Wh + (size_t)n * HID;  // B[k][n] = W_hh[n][k]

  for (int t = 0; t < SEQ; ++t) {
    int tp = dir ? (SEQ - 1 - t) : t;
    v8f acc0 = (v8f)0.0f, acc1 = (v8f)0.0f;  // batches 0..15 / 16..31
    for (int k = 0; k < HID; k += 4) {
      v2f bv = *(const v2f*)(brow + k + koff);
      v2f a0 = *(const v2f*)(&h_lds[lm][k + koff]);
      v2f a1 = *(const v2f*)(&h_lds[16 + lm][k + koff]);
      acc0 = __builtin_amdgcn_wmma_f32_16x16x4_f32(false, a0, false, bv,
                                                   (short)0, acc0, false, false);
      acc1 = __builtin_amdgcn_wmma_f32_16x16x4_f32(false, a1, false, bv,
                                                   (short)0, acc1, false, false);
    }
    int rb = (lane >> 4) << 3;
#pragma unroll
    for (int i = 0; i < 8; ++i) {
      g_lds[rb + i][n]      = acc0[i];
      g_lds[16 + rb + i][n] = acc1[i];
    }
    __syncthreads();

    const float* xrow = xgd + (size_t)tp * BATCH * GATES;
#pragma unroll
    for (int r = 0; r < 4; ++r) {
      int p = tid + 1024 * r;
      int b = p >> 7, j = p & 127;
      float ig = g_lds[b][j]         + xrow[b * GATES + j];
      float fg = g_lds[b][128 + j]   + xrow[b * GATES + 128 + j];
      float gg = g_lds[b][256 + j]   + xrow[b * GATES + 256 + j];
      float og = g_lds[b][384 + j]   + xrow[b * GATES + 384 + j];
      float c  = sigm(fg) * creg[r] + sigm(ig) * tanhf(gg);
      creg[r]  = c;
      float hn = sigm(og) * tanhf(c);
      h_lds[b][j] = hn;
      hs[(size_t)(tp * BATCH + b) * (2 * HID) + dir * HID + j] = hn;
    }
    __syncthreads();
  }
}

// ---------------------------------------------------------------------------
// Kernel C1: feats[t*32+b][k] = hs_row . W_out[k] + b_out[k]   (tiny GEMM)
// ---------------------------------------------------------------------------
__global__ __launch_bounds__(256) void k_feats(
    const float* __restrict__ hs, const float* __restrict__ Wout,
    const float* __restrict__ bout, float* __restrict__ feats)
{
  int idx = blockIdx.x * 256 + threadIdx.x;
  if (idx >= SEQ * BATCH * NTAG) return;
  int row = idx / NTAG, k = idx % NTAG;
  const float* hrow = hs + (size_t)row * (2 * HID);
  const float* w = Wout + (size_t)k * (2 * HID);
  float s = bout[k];
  for (int h = 0; h < 2 * HID; ++h) s += hrow[h] * w[h];
  feats[idx] = s;
}

// ---------------------------------------------------------------------------
// Kernel C2: Viterbi max-plus scan + LDS-resident packed backtrace.
// 1 block, 32x12 threads; pointers packed 2 tags/byte in LDS (98KB) so the
// serial backtrace is an LDS chain, not 512 dependent global loads.
// ---------------------------------------------------------------------------
__global__ __launch_bounds__(384) void k_viterbi(
    const float* __restrict__ feats, const float* __restrict__ trans,
    int* __restrict__ out)
{
  __shared__ unsigned char ptrs[SEQ][BATCH][NTAG / 2];  // 98304 B
  __shared__ unsigned char stage[BATCH][NTAG];
  __shared__ float sc[2][BATCH][NTAG];
  __shared__ float fin[BATCH][NTAG];
  __shared__ int best[BATCH];

  int tid = threadIdx.x;
  int b = tid / NTAG, k = tid % NTAG;       // k = next-tag
  float tr[NTAG];
#pragma unroll
  for (int p = 0; p < NTAG; ++p) tr[p] = trans[k * NTAG + p];
  sc[0][b][k] = (k == TAG_START) ? 0.0f : NEGV;
  __syncthreads();

  for (int t = 0; t < SEQ; ++t) {
    int cur = t & 1, nxt = cur ^ 1;
    float m = sc[cur][b][0] + tr[0];
    int am = 0;
#pragma unroll
    for (int p = 1; p < NTAG; ++p) {
      float v = sc[cur][b][p] + tr[p];
      if (v > m) { m = v; am = p; }         // strict > => first argmax (jnp semantics)
    }
    sc[nxt][b][k] = m + feats[(size_t)(t * BATCH + b) * NTAG + k];
    stage[b][k] = (unsigned char)am;
    __syncthreads();
    if (k < NTAG / 2)
      ptrs[t][b][k] = (unsigned char)(stage[b][2 * k] | (stage[b][2 * k + 1] << 4));
    __syncthreads();
  }
  // SEQ=512 even -> last write landed in buffer 0
  fin[b][k] = sc[0][b][k] + trans[TAG_STOP * NTAG + k];
  __syncthreads();
  if (k == 0) {
    float m = fin[b][0]; int am = 0;
    for (int p = 1; p < NTAG; ++p)
      if (fin[b][p] > m) { m = fin[b][p]; am = p; }
    best[b] = am;
  }
  __syncthreads();
  if (tid < BATCH) {
    int bb = tid;
    int tag = best[bb];
    for (int t = SEQ - 1; t >= 0; --t) {
      out[bb * SEQ + t] = tag;
      unsigned char byte = ptrs[t][bb][tag >> 1];
      tag = (tag & 1) ? (byte >> 4) : (byte & 15);
    }
  }
}

// ---------------------------------------------------------------------------
extern "C" void kernel_launch(void* const* d_in, const int* in_sizes, int n_in,
                              void* d_out, int out_size, void* d_ws, size_t ws_size,
                              hipStream_t stream) {
  const int*   sentence = (const int*)d_in[0];
  const float* h0    = (const float*)d_in[1];
  const float* c0    = (const float*)d_in[2];
  const float* embed = (const float*)d_in[3];
  const float* Wihf  = (const float*)d_in[4];
  const float* Whhf  = (const float*)d_in[5];
  const float* bf    = (const float*)d_in[6];
  const float* Wihr  = (const float*)d_in[7];
  const float* Whhr  = (const float*)d_in[8];
  const float* br    = (const float*)d_in[9];
  const float* Wout  = (const float*)d_in[10];
  const float* bout  = (const float*)d_in[11];
  const float* trans = (const float*)d_in[12];

  float* ws    = (float*)d_ws;
  float* xg    = ws;                                          // 2*16384*512 f32 (64 MB)
  float* hs    = xg + (size_t)2 * SEQ * BATCH * GATES;        // 16384*256 f32 (16 MB)
  float* feats = hs + (size_t)SEQ * BATCH * 2 * HID;          // 16384*12 f32
  int*   out   = (int*)d_out;

  k_input_gemm<<<8192, 256, 0, stream>>>(sentence, embed, Wihf, bf, Wihr, br, xg);
  k_lstm<<<2, 1024, 0, stream>>>(Whhf, Whhr, h0, c0, xg, hs);
  k_feats<<<(SEQ * BATCH * NTAG + 255) / 256, 256, 0, stream>>>(hs, Wout, bout, feats);
  k_viterbi<<<1, 384, 0, stream>>>(feats, trans, out);
}